// CompGCNBase_13503377179026
// MI455X (gfx1250) — compile-verified
//
#include <hip/hip_runtime.h>
#include <hip/hip_bf16.h>
#include <math.h>

// Problem sizes (from reference)
#define NUM_ENT  50000
#define NUM_REL  200
#define DD       200
#define NUM_EVT  20000
#define ROLE_NUM 20
#define RT_DIM   100
#define R_SLOTS  8
#define K_SLOTS  8
#define N_EDGES  800000

typedef __attribute__((ext_vector_type(2))) float v2f;
typedef __attribute__((ext_vector_type(8))) float v8f;

// ---------------------------------------------------------------------------
// fp32 WMMA 16x16x4 (exact fp32 math on the matrix pipe).
// A fragment (16x4, ISA 7.12.2): lane<16 -> row=lane, holds K=0,1 in .x/.y;
//                                lane>=16 -> row=lane-16, holds K=2,3.
// B fragment (4x16): lane<16 -> col=lane, rows K=0(.x),1(.y);
//                    lane>=16 -> col=lane-16, rows K=2(.x),3(.y).
// C/D (16x16): vgpr g: lanes0-15 -> M=g, lanes16-31 -> M=g+8; N=lane&15.
// ---------------------------------------------------------------------------
__device__ __forceinline__ v8f wmma_f32x4(v2f a, v2f b, v8f c) {
#if __has_builtin(__builtin_amdgcn_wmma_f32_16x16x4_f32)
  return __builtin_amdgcn_wmma_f32_16x16x4_f32(
      /*neg_a=*/false, a, /*neg_b=*/false, b,
      /*c_mod=*/(short)0, c, /*reuse_a=*/false, /*reuse_b=*/false);
#else
  int lane = threadIdx.x & 31;
  int n = lane & 15;
  float B0 = __shfl(b.x, n, 32),      B1 = __shfl(b.y, n, 32);
  float B2 = __shfl(b.x, n + 16, 32), B3 = __shfl(b.y, n + 16, 32);
  int mh = (lane >> 4) << 3;
#pragma unroll
  for (int g = 0; g < 8; ++g) {
    int m = mh + g;
    float A0 = __shfl(a.x, m, 32),      A1 = __shfl(a.y, m, 32);
    float A2 = __shfl(a.x, m + 16, 32), A3 = __shfl(a.y, m + 16, 32);
    c[g] += A0 * B0 + A1 * B1 + A2 * B2 + A3 * B3;
  }
  return c;
#endif
}

// ---------------------------------------------------------------------------
// Row-major fp32 GEMM: C = f(A@B [+ A2@B2] [+ addend]), f = tanh optional.
// Block = 8 waves sharing one 16-wide N tile; B panels staged in LDS
// column-major (pitch 202/102 -> aligned ds_load_b64, spread banks, zero-
// padded for N>200). Inner loop: global_load_b64 (A) + ds_load_b64 (B) + WMMA,
// no predication. No early exit: EXEC stays all-1s for WMMA and __syncthreads.
// M must be a multiple of 16 (20000 / 50000 both are); N bound-checked.
// ---------------------------------------------------------------------------
#define GEMM_WAVES 8
#define KP1 202   // LDS pitch for B panel (K <= 200)
#define KP2 102   // LDS pitch for B2 panel (K2 <= 100)

__global__ __launch_bounds__(32 * GEMM_WAVES)
void k_gemm_wmma(const float* __restrict__ A, int lda,
                 const float* __restrict__ B, int ldb,
                 const float* __restrict__ A2, int lda2,
                 const float* __restrict__ B2, int ldb2,
                 const float* __restrict__ addend,
                 float* __restrict__ C, int ldc,
                 int M, int N, int K, int K2, int dotanh) {
  __shared__ float sB[16 * KP1];
  __shared__ float sB2[16 * KP2];
  const int tn = blockIdx.x;

  // Stage B column panel (cols tn*16 .. tn*16+15), zero-padded past N.
  for (int idx = threadIdx.x; idx < K * 16; idx += blockDim.x) {
    int c = idx & 15, k = idx >> 4;
    int n = tn * 16 + c;
    sB[c * KP1 + k] = (n < N) ? B[(size_t)k * ldb + n] : 0.f;
  }
  if (A2) {
    for (int idx = threadIdx.x; idx < K2 * 16; idx += blockDim.x) {
      int c = idx & 15, k = idx >> 4;
      int n = tn * 16 + c;
      sB2[c * KP2 + k] = (n < N) ? B2[(size_t)k * ldb2 + n] : 0.f;
    }
  }
  __syncthreads();

  const int lane = threadIdx.x & 31;
  const int wv   = threadIdx.x >> 5;
  const int tm   = blockIdx.y * GEMM_WAVES + wv;     // wave-uniform
  const int Mtiles = M >> 4;
  const int tmc  = (tm < Mtiles) ? tm : (Mtiles - 1); // clamp: loads stay in-bounds
  const int rr = lane & 15;
  const int kh = (lane >> 4) << 1;                    // 0 or 2

  v8f acc = {};
  {
    const float* __restrict__ Ar = A + (size_t)(tmc * 16 + rr) * lda;
    const float* __restrict__ Bc = sB + rr * KP1;
    for (int k = 0; k < K; k += 4) {
      v2f a, b;
      a.x = Ar[k + kh];  a.y = Ar[k + kh + 1];
      b.x = Bc[k + kh];  b.y = Bc[k + kh + 1];
      acc = wmma_f32x4(a, b, acc);
    }
  }
  if (A2) {
    const float* __restrict__ Ar = A2 + (size_t)(tmc * 16 + rr) * lda2;
    const float* __restrict__ Bc = sB2 + rr * KP2;
    for (int k = 0; k < K2; k += 4) {
      v2f a, b;
      a.x = Ar[k + kh];  a.y = Ar[k + kh + 1];
      b.x = Bc[k + kh];  b.y = Bc[k + kh + 1];
      acc = wmma_f32x4(a, b, acc);
    }
  }

  const int n = tn * 16 + rr;
  if (tm < Mtiles && n < N) {
    const int mbase = tm * 16 + ((lane >> 4) << 3);
#pragma unroll
    for (int g = 0; g < 8; ++g) {
      size_t o = (size_t)(mbase + g) * ldc + n;
      float v = acc[g];
      if (addend) v += addend[o];
      if (dotanh) v = tanhf(v);
      C[o] = v;
    }
  }
}

// ---------------------------------------------------------------------------
// Small helper kernels
// ---------------------------------------------------------------------------
__global__ void k_zero(float* p, long n) {
  long i = (long)blockIdx.x * blockDim.x + threadIdx.x;
  long s = (long)gridDim.x * blockDim.x;
  for (; i < n; i += s) p[i] = 0.f;
}

__global__ void k_naive_gemm(const float* __restrict__ A, const float* __restrict__ B,
                             float* __restrict__ C, int M, int N, int K) {
  int idx = blockIdx.x * blockDim.x + threadIdx.x;
  if (idx >= M * N) return;
  int m = idx / N, n = idx % N;
  float s = 0.f;
  for (int k = 0; k < K; ++k) s += A[(size_t)m * K + k] * B[(size_t)k * N + n];
  C[idx] = s;
}

// ev_lin[e,d] = masked-mean over role slots of init_embed[ent]*role_proj[rt] + event_embed[e,d]
__global__ void k_event_pre(const float* __restrict__ init_embed,
                            const float* __restrict__ role_proj,
                            const float* __restrict__ event_embed,
                            const int* __restrict__ event_index,
                            const int* __restrict__ role_type,
                            const int* __restrict__ role_mask,
                            float* __restrict__ ev_lin) {
  int idx = blockIdx.x * blockDim.x + threadIdx.x;
  if (idx >= NUM_EVT * DD) return;
  int e = idx / DD, d = idx % DD;
  float s = 0.f, cnt = 0.f;
#pragma unroll
  for (int r = 0; r < R_SLOTS; ++r) {
    int m = role_mask[r * NUM_EVT + e];
    if (m) {
      int ent = event_index[r * NUM_EVT + e];
      int rt  = role_type[r * NUM_EVT + e];
      s += init_embed[(size_t)ent * DD + d] * role_proj[(size_t)rt * DD + d];
      cnt += 1.f;
    }
  }
  ev_lin[idx] = s / fmaxf(cnt, 1.f) + event_embed[idx];
}

// ent_agg[i,d] = masked-mean over event slots of ev_repr[evt]
__global__ void k_entity_agg(const float* __restrict__ ev_repr,
                             const int* __restrict__ entity_event_index,
                             const int* __restrict__ entity_mask,
                             float* __restrict__ ent_agg) {
  long idx = (long)blockIdx.x * blockDim.x + threadIdx.x;
  if (idx >= (long)NUM_ENT * DD) return;
  int i = (int)(idx / DD), d = (int)(idx % DD);
  float s = 0.f, cnt = 0.f;
#pragma unroll
  for (int k = 0; k < K_SLOTS; ++k) {
    int m = entity_mask[k * NUM_ENT + i];
    if (m) {
      int ev = entity_event_index[k * NUM_ENT + i];
      s += ev_repr[(size_t)ev * DD + d];
      cnt += 1.f;
    }
  }
  ent_agg[idx] = s / fmaxf(cnt, 1.f);
}

// One wave per edge: sum[dst] += xW[src] - relW[et]; cnt[dst] += 1
__global__ void k_edge_scatter(const float* __restrict__ xW,
                               const float* __restrict__ relW,
                               const int* __restrict__ src,
                               const int* __restrict__ dst,
                               const int* __restrict__ et,
                               int nEdges,
                               float* __restrict__ sum,
                               float* __restrict__ cnt) {
  int edge = blockIdx.x * (blockDim.x >> 5) + (threadIdx.x >> 5);
  if (edge >= nEdges) return;
  int lane = threadIdx.x & 31;
  int s = src[edge], de = dst[edge], t = et[edge];
  const float* __restrict__ xs = xW + (size_t)s * DD;
  const float* __restrict__ rs = relW + (size_t)t * DD;
  float* __restrict__ o = sum + (size_t)de * DD;
  for (int d = lane; d < DD; d += 32) atomicAdd(&o[d], xs[d] - rs[d]);
  if (lane == 0) atomicAdd(&cnt[de], 1.0f);
}

// out = tanh((in_mean + out_mean + (xW_loop - loopW)) / 3 + bias)
__global__ void k_finalize(const float* __restrict__ in_sum,
                           const float* __restrict__ out_sum,
                           const float* __restrict__ in_cnt,
                           const float* __restrict__ out_cnt,
                           const float* __restrict__ xW_loop,
                           const float* __restrict__ loopW,
                           const float* __restrict__ bias,
                           float* __restrict__ out) {
  long idx = (long)blockIdx.x * blockDim.x + threadIdx.x;
  if (idx >= (long)NUM_ENT * DD) return;
  int i = (int)(idx / DD), d = (int)(idx % DD);
  float iv = in_sum[idx]  / fmaxf(in_cnt[i], 1.f);
  float ov = out_sum[idx] / fmaxf(out_cnt[i], 1.f);
  float lv = xW_loop[idx] - loopW[d];
  out[idx] = tanhf((iv + ov + lv) * (1.0f / 3.0f) + bias[d]);
}

// ---------------------------------------------------------------------------
extern "C" void kernel_launch(void* const* d_in, const int* in_sizes, int n_in,
                              void* d_out, int out_size, void* d_ws, size_t ws_size,
                              hipStream_t stream) {
  const float* init_embed      = (const float*)d_in[0];
  const float* rel_embed       = (const float*)d_in[1];
  const float* role_type_embed = (const float*)d_in[2];
  const float* event_embed     = (const float*)d_in[3];
  const float* evt_type_emb    = (const float*)d_in[4];
  const float* W_role          = (const float*)d_in[5];
  const float* W_type          = (const float*)d_in[6];
  const float* W_event         = (const float*)d_in[7];
  const float* W_self          = (const float*)d_in[8];
  const float* w_in            = (const float*)d_in[9];
  const float* w_out           = (const float*)d_in[10];
  const float* w_loop          = (const float*)d_in[11];
  const float* w_rel            = (const float*)d_in[12];
  const float* loop_rel        = (const float*)d_in[13];
  const float* bias            = (const float*)d_in[14];
  const int* event_index        = (const int*)d_in[15];
  const int* role_type          = (const int*)d_in[16];
  const int* role_mask          = (const int*)d_in[17];
  const int* entity_event_index = (const int*)d_in[18];
  const int* entity_mask        = (const int*)d_in[19];
  const int* edge_index         = (const int*)d_in[20];
  const int* edge_type          = (const int*)d_in[21];

  float* out_x = (float*)d_out;
  float* out_r = out_x + (size_t)NUM_ENT * DD;

  // Bump-allocate workspace (floats)
  float* ws = (float*)d_ws;
  size_t off = 0;
  auto alloc = [&](size_t n) { float* p = ws + off; off += n; return p; };
  float* role_proj = alloc((size_t)ROLE_NUM * DD);       // 20x200
  float* relW_in   = alloc((size_t)2 * NUM_REL * DD);    // 400x200
  float* relW_out  = alloc((size_t)2 * NUM_REL * DD);    // 400x200
  float* WT_WE     = alloc((size_t)RT_DIM * DD);         // 100x200 = W_type@W_event
  float* loopW     = alloc((size_t)DD);                  // 1x200 = loop_rel@w_loop
  float* in_cnt    = alloc((size_t)NUM_ENT);
  float* out_cnt   = alloc((size_t)NUM_ENT);             // contiguous with in_cnt
  float* ev_lin    = alloc((size_t)NUM_EVT * DD);
  float* ev_repr   = alloc((size_t)NUM_EVT * DD);
  float* ent_agg   = alloc((size_t)NUM_ENT * DD);
  float* x         = alloc((size_t)NUM_ENT * DD);
  float* xW_in     = alloc((size_t)NUM_ENT * DD);
  float* xW_out    = alloc((size_t)NUM_ENT * DD);
  float* xW_loop   = alloc((size_t)NUM_ENT * DD);
  float* in_sum    = alloc((size_t)NUM_ENT * DD);
  float* out_sum   = alloc((size_t)NUM_ENT * DD);        // contiguous with in_sum

  const int TB = 256;
  // ---- tiny precomputed GEMMs (scalar; negligible work) ----
  k_naive_gemm<<<(ROLE_NUM * DD + TB - 1) / TB, TB, 0, stream>>>(role_type_embed, W_role, role_proj, ROLE_NUM, DD, RT_DIM);
  k_naive_gemm<<<(2 * NUM_REL * DD + TB - 1) / TB, TB, 0, stream>>>(rel_embed, w_in,  relW_in,  2 * NUM_REL, DD, DD);
  k_naive_gemm<<<(2 * NUM_REL * DD + TB - 1) / TB, TB, 0, stream>>>(rel_embed, w_out, relW_out, 2 * NUM_REL, DD, DD);
  k_naive_gemm<<<(RT_DIM * DD + TB - 1) / TB, TB, 0, stream>>>(W_type, W_event, WT_WE, RT_DIM, DD, DD);
  k_naive_gemm<<<(DD + TB - 1) / TB, TB, 0, stream>>>(loop_rel, w_loop, loopW, 1, DD, DD);
  k_naive_gemm<<<(2 * NUM_REL * DD + TB - 1) / TB, TB, 0, stream>>>(rel_embed, w_rel, out_r, 2 * NUM_REL, DD, DD);

  // ---- EventConv ----
  k_event_pre<<<(NUM_EVT * DD + TB - 1) / TB, TB, 0, stream>>>(
      init_embed, role_proj, event_embed, event_index, role_type, role_mask, ev_lin);

  const int NT = (DD + 15) / 16;  // 13 column tiles
  {
    dim3 g(NT, (NUM_EVT / 16 + GEMM_WAVES - 1) / GEMM_WAVES);
    // ev_repr = tanh(ev_lin @ W_event + evt_type_emb @ (W_type@W_event))
    k_gemm_wmma<<<g, 32 * GEMM_WAVES, 0, stream>>>(
        ev_lin, DD, W_event, DD, evt_type_emb, RT_DIM, WT_WE, DD,
        nullptr, ev_repr, DD, NUM_EVT, DD, DD, RT_DIM, 1);
  }

  k_entity_agg<<<(int)(((long)NUM_ENT * DD + TB - 1) / TB), TB, 0, stream>>>(
      ev_repr, entity_event_index, entity_mask, ent_agg);

  dim3 gEnt(NT, (NUM_ENT / 16 + GEMM_WAVES - 1) / GEMM_WAVES);
  // x = tanh(init_embed @ W_self + ent_agg)
  k_gemm_wmma<<<gEnt, 32 * GEMM_WAVES, 0, stream>>>(
      init_embed, DD, W_self, DD, nullptr, 0, nullptr, 0,
      ent_agg, x, DD, NUM_ENT, DD, DD, 0, 1);

  // ---- CompGCN: node-level transforms (edge GEMM factored out) ----
  k_gemm_wmma<<<gEnt, 32 * GEMM_WAVES, 0, stream>>>(
      x, DD, w_in, DD, nullptr, 0, nullptr, 0, nullptr, xW_in, DD, NUM_ENT, DD, DD, 0, 0);
  k_gemm_wmma<<<gEnt, 32 * GEMM_WAVES, 0, stream>>>(
      x, DD, w_out, DD, nullptr, 0, nullptr, 0, nullptr, xW_out, DD, NUM_ENT, DD, DD, 0, 0);
  k_gemm_wmma<<<gEnt, 32 * GEMM_WAVES, 0, stream>>>(
      x, DD, w_loop, DD, nullptr, 0, nullptr, 0, nullptr, xW_loop, DD, NUM_ENT, DD, DD, 0, 0);

  // ---- edge phase ----
  k_zero<<<1024, TB, 0, stream>>>(in_cnt, (long)2 * NUM_ENT);           // in_cnt + out_cnt
  k_zero<<<8192, TB, 0, stream>>>(in_sum, (long)2 * NUM_ENT * DD);      // in_sum + out_sum

  const int half = N_EDGES / 2;
  const int* src = edge_index;
  const int* dst = edge_index + N_EDGES;
  const int edgesPerBlock = TB / 32;
  k_edge_scatter<<<(half + edgesPerBlock - 1) / edgesPerBlock, TB, 0, stream>>>(
      xW_in, relW_in, src, dst, edge_type, half, in_sum, in_cnt);
  k_edge_scatter<<<(half + edgesPerBlock - 1) / edgesPerBlock, TB, 0, stream>>>(
      xW_out, relW_out, src + half, dst + half, edge_type + half, half, out_sum, out_cnt);

  // ---- finalize ----
  k_finalize<<<(int)(((long)NUM_ENT * DD + TB - 1) / TB), TB, 0, stream>>>(
      in_sum, out_sum, in_cnt, out_cnt, xW_loop, loopW, bias, out_x);

  (void)in_sizes; (void)n_in; (void)out_size; (void)ws_size;
}